// HeteroGATEncoder_48876727828799
// MI455X (gfx1250) — compile-verified
//
#include <hip/hip_runtime.h>

#define DD 128      // embedding dim
#define HH 4        // heads
#define CC 32       // channels per head
#define NEG_SLOPE 0.2f

typedef __attribute__((ext_vector_type(2))) float v2f;
typedef __attribute__((ext_vector_type(8))) float v8f;

// ---------------------------------------------------------------------------
// GEMM: Out[N,128] = X[N,128] @ W[128,128]   (fp32, V_WMMA_F32_16X16X4_F32)
// Block = 256 threads = 8 waves. Block covers 16 rows x 128 cols of Out.
// Wave w covers column tile [16w, 16w+16). K loop: 32 steps of K=4.
// A fragment (16x4 f32): lane l: M=l&15, K pair base=(l>>4)*2, vgpr j -> K+j
// B fragment (4x16 f32): lane l: N=l&15, same K mapping
// C/D (16x16 f32): vgpr j, lane l -> M = j + 8*(l>>4), N = l&15
// ---------------------------------------------------------------------------
__global__ __launch_bounds__(256) void gemm_xw_wmma(
    const float* __restrict__ X, const float* __restrict__ W,
    float* __restrict__ Out, int N)
{
  __shared__ float As[16][132];   // padded: stride 132 -> conflict-free column reads
  const int row0 = blockIdx.x * 16;
  const int tid  = threadIdx.x;

  // cooperative load of the 16x128 A tile (each thread: 8 contiguous floats)
  {
    const int r  = tid >> 4;
    const int c0 = (tid & 15) * 8;
    const float4* src = (const float4*)(X + (size_t)(row0 + r) * DD + c0);
    float4 a0 = src[0];
    float4 a1 = src[1];
    float* dstp = &As[r][c0];
    dstp[0] = a0.x; dstp[1] = a0.y; dstp[2] = a0.z; dstp[3] = a0.w;
    dstp[4] = a1.x; dstp[5] = a1.y; dstp[6] = a1.z; dstp[7] = a1.w;
  }
  __syncthreads();

  const int wave = tid >> 5;           // 0..7 column tile
  const int lane = tid & 31;
  const int mn   = lane & 15;
  const int kh   = (lane >> 4) << 1;   // 0 or 2
  const int col  = wave * 16 + mn;

  v8f acc = {};
  #pragma unroll
  for (int kb = 0; kb < DD; kb += 4) {
    v2f a, b;
    a.x = As[mn][kb + kh];
    a.y = As[mn][kb + kh + 1];
    b.x = W[(size_t)(kb + kh)     * DD + col];
    b.y = W[(size_t)(kb + kh + 1) * DD + col];
    acc = __builtin_amdgcn_wmma_f32_16x16x4_f32(
        false, a, false, b, (short)0, acc, false, false);
  }

  const int rbase = row0 + ((lane >> 4) << 3);
  #pragma unroll
  for (int j = 0; j < 8; ++j)
    Out[(size_t)(rbase + j) * DD + col] = acc[j];
}

// ---------------------------------------------------------------------------
// alpha[n,h] = sum_c h[n, h*32 + c] * a[h, c]
// ---------------------------------------------------------------------------
__global__ __launch_bounds__(256) void attn_alpha_kernel(
    const float* __restrict__ h, const float* __restrict__ a,
    float* __restrict__ out, int N)
{
  int i = blockIdx.x * blockDim.x + threadIdx.x;
  if (i >= N * HH) return;
  int n = i >> 2, hh = i & 3;
  const float4* hp = (const float4*)(h + (size_t)n * DD + hh * CC);
  const float4* ap = (const float4*)(a + hh * CC);
  float s = 0.f;
  #pragma unroll
  for (int j = 0; j < 8; ++j) {
    float4 x = hp[j], y = ap[j];
    s += x.x * y.x + x.y * y.y + x.z * y.z + x.w * y.w;
  }
  out[i] = s;
}

// order-preserving float <-> uint mapping for atomic segment-max
__device__ __forceinline__ unsigned f2ord(float f) {
  unsigned u = __float_as_uint(f);
  return (u & 0x80000000u) ? ~u : (u | 0x80000000u);
}
__device__ __forceinline__ float ord2f(unsigned u) {
  unsigned b = (u & 0x80000000u) ? (u & 0x7FFFFFFFu) : ~u;
  return __uint_as_float(b);
}
__device__ __forceinline__ float leaky(float v) {
  return v > 0.f ? v : NEG_SLOPE * v;
}

// pass 1: per-destination segment max of leakyrelu(as[src]+ad[dst])
__global__ __launch_bounds__(256) void edge_max_kernel(
    const int* __restrict__ ei, const float* __restrict__ as,
    const float* __restrict__ ad, unsigned* __restrict__ emax, int E)
{
  int e = blockIdx.x * blockDim.x + threadIdx.x;
  if (e >= E) return;
  int s = ei[e], d = ei[E + e];
  #pragma unroll
  for (int hh = 0; hh < HH; ++hh) {
    float v = leaky(as[s * HH + hh] + ad[d * HH + hh]);
    atomicMax(&emax[d * HH + hh], f2ord(v));
  }
}

// pass 2: ex = exp(e - emax[dst]); denom[dst] += ex; cache ex per edge
__global__ __launch_bounds__(256) void edge_expsum_kernel(
    const int* __restrict__ ei, const float* __restrict__ as,
    const float* __restrict__ ad, const unsigned* __restrict__ emax,
    float* __restrict__ denom, float* __restrict__ exbuf, int E)
{
  int e = blockIdx.x * blockDim.x + threadIdx.x;
  if (e >= E) return;
  int s = ei[e], d = ei[E + e];
  #pragma unroll
  for (int hh = 0; hh < HH; ++hh) {
    float v = leaky(as[s * HH + hh] + ad[d * HH + hh]);
    float m = ord2f(emax[d * HH + hh]);
    // reference: non-finite segment max -> 0
    if ((__float_as_uint(m) & 0x7F800000u) == 0x7F800000u) m = 0.f;
    float ex = expf(v - m);
    exbuf[(size_t)e * HH + hh] = ex;
    atomicAdd(&denom[d * HH + hh], ex);
  }
}

// pass 3: acc[dst] += hs[src] * (ex / (denom[dst] + 1e-16)); one wave per edge
__global__ __launch_bounds__(256) void edge_scatter_kernel(
    const int* __restrict__ ei, const float* __restrict__ hs,
    const float* __restrict__ exbuf, const float* __restrict__ denom,
    float* __restrict__ acc, int E)
{
  int gw   = (blockIdx.x * blockDim.x + threadIdx.x) >> 5;  // edge id
  int lane = threadIdx.x & 31;
  if (gw >= E) return;
  int s  = ei[gw], d = ei[E + gw];
  int hh = lane >> 3;  // channels [4*lane, 4*lane+4) all in head lane/8
  float alpha = exbuf[(size_t)gw * HH + hh] / (denom[d * HH + hh] + 1e-16f);
  float4 hv = ((const float4*)(hs + (size_t)s * DD))[lane];
  float* o = acc + (size_t)d * DD + lane * 4;
  atomicAdd(o + 0, hv.x * alpha);
  atomicAdd(o + 1, hv.y * alpha);
  atomicAdd(o + 2, hv.z * alpha);
  atomicAdd(o + 3, hv.w * alpha);
}

// finalize: h = elu(acc + bias), in place (float4 granularity)
__global__ __launch_bounds__(256) void finalize_kernel(
    float* __restrict__ acc, const float* __restrict__ bias, int nvec4)
{
  int i = blockIdx.x * blockDim.x + threadIdx.x;
  if (i >= nvec4) return;
  float4 v = ((float4*)acc)[i];
  float4 b = ((const float4*)bias)[i & (DD / 4 - 1)];
  v.x += b.x; v.y += b.y; v.z += b.z; v.w += b.w;
  v.x = v.x > 0.f ? v.x : expm1f(v.x);
  v.y = v.y > 0.f ? v.y : expm1f(v.y);
  v.z = v.z > 0.f ? v.z : expm1f(v.z);
  v.w = v.w > 0.f ? v.w : expm1f(v.w);
  ((float4*)acc)[i] = v;
}

// ---------------------------------------------------------------------------
// host orchestration
// ---------------------------------------------------------------------------
struct RelScratch {
  float* hs; float* hd; float* as; float* ad;
  unsigned* emax; float* denom; float* exbuf;
};

static void run_relation(const float* xsrc, const float* xdst,
                         const float* Wsrc, const float* Wdst,
                         const float* a_src, const float* a_dst,
                         const float* bias, const int* ei,
                         float* out, int N, int E,
                         const RelScratch& rs, hipStream_t stream)
{
  const size_t F = (size_t)N * DD;
  int gemm_blocks = (N + 15) / 16;
  gemm_xw_wmma<<<gemm_blocks, 256, 0, stream>>>(xsrc, Wsrc, rs.hs, N);
  gemm_xw_wmma<<<gemm_blocks, 256, 0, stream>>>(xdst, Wdst, rs.hd, N);

  int ablk = (N * HH + 255) / 256;
  attn_alpha_kernel<<<ablk, 256, 0, stream>>>(rs.hs, a_src, rs.as, N);
  attn_alpha_kernel<<<ablk, 256, 0, stream>>>(rs.hd, a_dst, rs.ad, N);

  hipMemsetAsync(rs.emax, 0, (size_t)N * HH * sizeof(unsigned), stream);
  hipMemsetAsync(rs.denom, 0, (size_t)N * HH * sizeof(float), stream);
  hipMemsetAsync(out, 0, F * sizeof(float), stream);

  int eblk = (E + 255) / 256;
  edge_max_kernel<<<eblk, 256, 0, stream>>>(ei, rs.as, rs.ad, rs.emax, E);
  edge_expsum_kernel<<<eblk, 256, 0, stream>>>(ei, rs.as, rs.ad, rs.emax,
                                               rs.denom, rs.exbuf, E);
  int sblk = (E * 32 + 255) / 256;
  edge_scatter_kernel<<<sblk, 256, 0, stream>>>(ei, rs.hs, rs.exbuf, rs.denom,
                                                out, E);
  int fblk = ((int)(F / 4) + 255) / 256;
  finalize_kernel<<<fblk, 256, 0, stream>>>(out, bias, (int)(F / 4));
}

extern "C" void kernel_launch(void* const* d_in, const int* in_sizes, int n_in,
                              void* d_out, int out_size, void* d_ws, size_t ws_size,
                              hipStream_t stream) {
  const float* x_food     = (const float*)d_in[0];
  const float* x_nut      = (const float*)d_in[1];
  const float* Wsrc_fn    = (const float*)d_in[2];
  const float* Wdst_fn    = (const float*)d_in[3];
  const float* att_src_fn = (const float*)d_in[4];
  const float* att_dst_fn = (const float*)d_in[5];
  const float* bias_fn    = (const float*)d_in[6];
  const float* Wsrc_nf    = (const float*)d_in[7];
  const float* Wdst_nf    = (const float*)d_in[8];
  const float* att_src_nf = (const float*)d_in[9];
  const float* att_dst_nf = (const float*)d_in[10];
  const float* bias_nf    = (const float*)d_in[11];
  const int*   ei_fn      = (const int*)d_in[12];
  const int*   ei_nf      = (const int*)d_in[13];

  const int N = in_sizes[0] / DD;        // 50000 (N_FOOD == N_NUT)
  const int E = in_sizes[12] / 2;        // 800000
  const size_t F = (size_t)N * DD;

  float* ws = (float*)d_ws;
  RelScratch rs;
  rs.hs      = ws;                       // F
  rs.hd      = ws + F;                   // F
  float* hb_food = ws + 2 * F;           // F (layer-0 food output)
  float* hb_nut  = ws + 3 * F;           // F (layer-0 nut output)
  rs.as      = ws + 4 * F;               // N*H
  rs.ad      = rs.as + (size_t)N * HH;   // N*H
  rs.emax    = (unsigned*)(rs.ad + (size_t)N * HH);        // N*H
  rs.denom   = (float*)rs.emax + (size_t)N * HH;           // N*H
  rs.exbuf   = rs.denom + (size_t)N * HH;                  // E*H

  float* out_food_l[2] = { hb_food, (float*)d_out };       // stack: [food, nut]
  float* out_nut_l[2]  = { hb_nut,  (float*)d_out + F };

  const float* cf = x_food;
  const float* cn = x_nut;
  for (int l = 0; l < 2; ++l) {
    // relation food -> nutrient (dst = nutrient)
    run_relation(cf, cn,
                 Wsrc_fn + (size_t)l * DD * DD, Wdst_fn + (size_t)l * DD * DD,
                 att_src_fn + (size_t)l * HH * CC, att_dst_fn + (size_t)l * HH * CC,
                 bias_fn + (size_t)l * DD, ei_fn, out_nut_l[l], N, E, rs, stream);
    // relation nutrient -> food (dst = food); uses pre-update h's
    run_relation(cn, cf,
                 Wsrc_nf + (size_t)l * DD * DD, Wdst_nf + (size_t)l * DD * DD,
                 att_src_nf + (size_t)l * HH * CC, att_dst_nf + (size_t)l * HH * CC,
                 bias_nf + (size_t)l * DD, ei_nf, out_food_l[l], N, E, rs, stream);
    cf = out_food_l[l];
    cn = out_nut_l[l];
  }
}